// OnlineTripletLoss_55860344652534
// MI455X (gfx1250) — compile-verified
//
#include <hip/hip_runtime.h>
#include <hip/hip_bf16.h>

#define B_N      512
#define D_DIM    128
#define MARGIN_F 0.2f
#define INF_F    1e30f

typedef __attribute__((ext_vector_type(2))) float v2f;
typedef __attribute__((ext_vector_type(8))) float v8f;

// ---------------------------------------------------------------------------
// Kernel 1: row norms + zero the global accumulators (must happen every call;
// graph replay does not re-poison d_ws).
// ---------------------------------------------------------------------------
__global__ void triplet_norms_init_kernel(const float* __restrict__ x,
                                          float* __restrict__ norms,
                                          float* __restrict__ acc) {
    int i = blockIdx.x * blockDim.x + threadIdx.x;
    if (i == 0) {
        acc[0] = 0.0f;            // running loss total
        ((int*)acc)[1] = 0;       // running triplet count
    }
    if (i < B_N) {
        const float* row = x + i * D_DIM;
        float s = 0.0f;
#pragma unroll 8
        for (int j = 0; j < D_DIM; ++j) {
            s = fmaf(row[j], row[j], s);
        }
        norms[i] = s;
    }
}

// ---------------------------------------------------------------------------
// Kernel 2: Gram matrix via V_WMMA_F32_16X16X4_F32, fused into clamped
// pairwise squared distances. One wave per 16x16 output tile; 32x32 tiles.
//
// f32 A 16x4 fragment:  lane l, VGPR v -> M = l&15, K = v + 2*(l>>4)
// f32 B 4x16 fragment:  lane l, VGPR v -> N = l&15, K = v + 2*(l>>4)
// f32 C/D 16x16:        lane l, VGPR v -> M = v + 8*(l>>4), N = l&15
// ---------------------------------------------------------------------------
__global__ void triplet_gram_wmma_kernel(const float* __restrict__ x,
                                         const float* __restrict__ norms,
                                         float* __restrict__ dsq) {
    const int wave = (blockIdx.x * blockDim.x + threadIdx.x) >> 5;
    const int lane = threadIdx.x & 31;
    const int tm   = wave >> 5;    // tile row   (0..31)
    const int tn   = wave & 31;    // tile col   (0..31)
    const int half = lane >> 4;    // 0: lanes 0-15, 1: lanes 16-31
    const int r    = lane & 15;

    // Both fragments load two consecutive K-values for K = k0 + 2*half.
    const float2* arow = (const float2*)(x + (size_t)(tm * 16 + r) * D_DIM);
    const float2* brow = (const float2*)(x + (size_t)(tn * 16 + r) * D_DIM);

    v8f acc = {0.f, 0.f, 0.f, 0.f, 0.f, 0.f, 0.f, 0.f};

#pragma unroll
    for (int k0 = 0; k0 < D_DIM; k0 += 4) {
        float2 av = arow[(k0 >> 1) + half];
        float2 bv = brow[(k0 >> 1) + half];
        v2f a = {av.x, av.y};
        v2f b = {bv.x, bv.y};
        // 8 args: (neg_a, A, neg_b, B, c_mod, C, reuse_a, reuse_b)
        acc = __builtin_amdgcn_wmma_f32_16x16x4_f32(
            false, a, false, b, (short)0, acc, false, false);
    }

    const int col = tn * 16 + r;
    const float nc = norms[col];
#pragma unroll
    for (int v = 0; v < 8; ++v) {
        const int row = tm * 16 + v + 8 * half;
        float d = norms[row] + nc - 2.0f * acc[v];
        dsq[(size_t)row * B_N + col] = fmaxf(d, 0.0f);
    }
}

// ---------------------------------------------------------------------------
// Kernel 3: semi-hard mining. One block per anchor; the anchor's distance row
// and the labels live in LDS; thread t is candidate positive p = t and scans
// all 512 negatives from LDS. Block-reduced, then one atomic pair per block.
// ---------------------------------------------------------------------------
__global__ void triplet_mine_kernel(const float* __restrict__ dsq,
                                    const int* __restrict__ labels,
                                    float* __restrict__ acc) {
    __shared__ float drow[B_N];
    __shared__ int   lab[B_N];
    __shared__ float s_total;
    __shared__ int   s_count;

    const int a = blockIdx.x;
    const int t = threadIdx.x;

    drow[t] = dsq[(size_t)a * B_N + t];
    lab[t]  = labels[t];
    if (t == 0) { s_total = 0.0f; s_count = 0; }
    __syncthreads();

    const int la = lab[a];
    float loss = 0.0f;
    int   cnt  = 0;

    if (t != a && lab[t] == la) {            // (a, p=t) is a positive pair
        const float d_ap = drow[t];
        const float thr  = d_ap + MARGIN_F;
        float min_semi  = INF_F;
        float min_valid = INF_F;
        bool  has_semi  = false;
        bool  has_valid = false;
        for (int n = 0; n < B_N; ++n) {
            const float d_an = drow[n];
            const bool gt = (lab[n] != la) && (d_an > d_ap);
            if (gt) {
                has_valid = true;
                min_valid = fminf(min_valid, d_an);
                if (d_an < thr) {
                    has_semi = true;
                    min_semi = fminf(min_semi, d_an);
                }
            }
        }
        if (has_valid) {                      // has_semi implies has_valid
            const float d_n = has_semi ? min_semi : min_valid;
            loss = fmaxf(d_ap - d_n + MARGIN_F, 0.0f);
            cnt  = 1;
        }
    }

    if (cnt) {
        atomicAdd(&s_total, loss);
        atomicAdd(&s_count, 1);
    }
    __syncthreads();
    if (t == 0 && s_count > 0) {
        atomicAdd(&acc[0], s_total);
        atomicAdd(((int*)acc) + 1, s_count);
    }
}

// ---------------------------------------------------------------------------
// Kernel 4: finalize -> d_out = { avg_loss, (float)count }
// ---------------------------------------------------------------------------
__global__ void triplet_finalize_kernel(const float* __restrict__ acc,
                                        float* __restrict__ out) {
    const float total = acc[0];
    const int   count = ((const int*)acc)[1];
    out[0] = (count > 0) ? (total / (float)count) : 0.0f;
    out[1] = (float)count;
}

// ---------------------------------------------------------------------------
extern "C" void kernel_launch(void* const* d_in, const int* in_sizes, int n_in,
                              void* d_out, int out_size, void* d_ws, size_t ws_size,
                              hipStream_t stream) {
    const float* x      = (const float*)d_in[0];   // embeddings [512, 128] f32
    const int*   labels = (const int*)d_in[1];     // labels [512] int
    float*       out    = (float*)d_out;           // {avg_loss, count}

    // Workspace layout (~1.05 MiB):
    float* dsq   = (float*)d_ws;            // 512*512 pairwise sq distances
    float* norms = dsq + (size_t)B_N * B_N; // 512 row norms
    float* acc   = norms + B_N;             // acc[0]=total (f32), acc[1]=count (i32)

    triplet_norms_init_kernel<<<2, 256, 0, stream>>>(x, norms, acc);
    triplet_gram_wmma_kernel<<<128, 256, 0, stream>>>(x, norms, dsq); // 1024 waves = 32x32 tiles
    triplet_mine_kernel<<<B_N, B_N, 0, stream>>>(dsq, labels, acc);
    triplet_finalize_kernel<<<1, 1, 0, stream>>>(acc, out);
}